// GAT_MTL_Model_26963804684694
// MI455X (gfx1250) — compile-verified
//
#include <hip/hip_runtime.h>
#include <hip/hip_bf16.h>

typedef __attribute__((ext_vector_type(2))) float v2f;
typedef __attribute__((ext_vector_type(8))) float v8f;

#define NNODES 20000
#define NEDGES 320000

#if defined(__AMDGCN__) && __has_builtin(__builtin_amdgcn_global_load_async_to_lds_b128) && \
    __has_builtin(__builtin_amdgcn_global_load_async_to_lds_b32) && \
    __has_builtin(__builtin_amdgcn_s_wait_asynccnt)
#define USE_ASYNC_LDS 1
#define AS1 __attribute__((address_space(1)))
#define AS3 __attribute__((address_space(3)))
typedef int vs4i __attribute__((vector_size(16)));   // matches builtin param type
#else
#define USE_ASYNC_LDS 0
#endif

// ---------------------------------------------------------------------------
// Tiled fp32 GEMM on the WMMA pipe: C[M,N] = A[M,K] @ B[K,N] (+bias)(+leaky 0.01)
// K % 32 == 0, N % 64 == 0. Block = 256 thr = 8 waves; tile 128x64, wave 16x64.
// Double-buffered LDS staging via GLOBAL_LOAD_ASYNC_TO_LDS (ASYNCcnt),
// B tile stored transposed so both WMMA operands are single ds_load_b64.
// ---------------------------------------------------------------------------
__device__ __forceinline__ void stage_tiles(
    const float* __restrict__ A, const float* __restrict__ B,
    int M, int N, int K, int kb, int m0, int n0, int t,
    float (*As)[36], float (*BsT)[36])
{
    // A tile 128x32: coalesced, one b128 per thread per row-group
#pragma unroll
    for (int r = 0; r < 4; ++r) {
        int row = r * 32 + (t >> 3);
        int col = (t & 7) * 4;
        int gr  = m0 + row;
        gr = gr < M ? gr : M - 1;            // clamp: garbage rows never stored
        const float* gp = A + (size_t)gr * K + kb + col;
#if USE_ASYNC_LDS
        __builtin_amdgcn_global_load_async_to_lds_b128(
            (AS1 vs4i*)gp, (AS3 vs4i*)&As[row][col], 0, 0);
#else
        *(float4*)&As[row][col] = *(const float4*)gp;
#endif
    }
    // B tile 32x64 -> stored transposed BsT[n][k]
#pragma unroll
    for (int r = 0; r < 2; ++r) {
        int row = r * 16 + (t >> 4);         // k index 0..31
        int col = (t & 15) * 4;              // n offset 0..60
        const float* gp = B + (size_t)(kb + row) * N + n0 + col;
#if USE_ASYNC_LDS
#pragma unroll
        for (int i = 0; i < 4; ++i)
            __builtin_amdgcn_global_load_async_to_lds_b32(
                (AS1 int*)(gp + i), (AS3 int*)&BsT[col + i][row], 0, 0);
#else
        float4 v = *(const float4*)gp;
        BsT[col + 0][row] = v.x;
        BsT[col + 1][row] = v.y;
        BsT[col + 2][row] = v.z;
        BsT[col + 3][row] = v.w;
#endif
    }
}

__device__ __forceinline__ void wait_stage()
{
#if USE_ASYNC_LDS
    __builtin_amdgcn_s_wait_asynccnt(0);
#endif
}

__global__ __launch_bounds__(256) void gemm_wmma_f32(
    const float* __restrict__ A, const float* __restrict__ B,
    const float* __restrict__ bias, float* __restrict__ C,
    int M, int N, int K, int act)
{
    __shared__ float As[2][128][36];   // pad 36: conflict-free stride, 8B-aligned pairs
    __shared__ float BsT[2][64][36];   // transposed B tile

    const int m0   = blockIdx.x * 128;
    const int n0   = blockIdx.y * 64;
    const int t    = threadIdx.x;
    const int wv   = t >> 5;
    const int lane = t & 31;
    const int hl   = lane >> 4;
    const int l15  = lane & 15;

    v8f acc[4];
#pragma unroll
    for (int j = 0; j < 4; ++j)
#pragma unroll
        for (int v = 0; v < 8; ++v) acc[j][v] = 0.0f;

    stage_tiles(A, B, M, N, K, 0, m0, n0, t, As[0], BsT[0]);
    wait_stage();
    __syncthreads();

    for (int kb = 0; kb < K; kb += 32) {
        const int cur  = (kb >> 5) & 1;
        const bool more = (kb + 32) < K;
        if (more)
            stage_tiles(A, B, M, N, K, kb + 32, m0, n0, t, As[cur ^ 1], BsT[cur ^ 1]);

        // Compute: both WMMA operands are contiguous 8B LDS pairs.
        const float* ap = &As[cur][wv * 16 + l15][2 * hl];
#pragma unroll
        for (int kk = 0; kk < 32; kk += 4) {
            v2f a = *(const v2f*)(ap + kk);
#pragma unroll
            for (int j = 0; j < 4; ++j) {
                v2f b = *(const v2f*)(&BsT[cur][j * 16 + l15][2 * hl] + kk);
                acc[j] = __builtin_amdgcn_wmma_f32_16x16x4_f32(
                    false, a, false, b, (short)0, acc[j], false, false);
            }
        }

        if (more) {
            wait_stage();
            __syncthreads();
        }
    }

    // C/D layout: VGPR v, lanes0-15 -> M=v, lanes16-31 -> M=v+8
#pragma unroll
    for (int j = 0; j < 4; ++j)
#pragma unroll
        for (int v = 0; v < 8; ++v) {
            int row = m0 + wv * 16 + v + 8 * hl;
            int col = n0 + j * 16 + l15;
            if (row < M) {
                float x = acc[j][v];
                if (bias) x += bias[col];
                if (act)  x = x > 0.f ? x : 0.01f * x;
                C[(size_t)row * N + col] = x;
            }
        }
}

// ---------------------------------------------------------------------------
// Attention scores: a_src[n,h] = dot(xl[n,h,:], att_src[h,:]) (and a_dst).
// One wave per (node, head); wave32 shuffle reduction.
// ---------------------------------------------------------------------------
__global__ __launch_bounds__(256) void att_scores_kernel(
    const float* __restrict__ xl, const float* __restrict__ att_src,
    const float* __restrict__ att_dst, float* __restrict__ asrc,
    float* __restrict__ adst, int npairs)
{
    int pair = (blockIdx.x * blockDim.x + threadIdx.x) >> 5;  // uniform per wave
    int lane = threadIdx.x & 31;
    if (pair >= npairs) return;
    int n = pair >> 2, h = pair & 3;
    const float* xr = xl + (size_t)n * 1024 + h * 256;
    const float* ar = att_src + h * 256;
    const float* br = att_dst + h * 256;
    float s1 = 0.f, s2 = 0.f;
    for (int i = lane; i < 256; i += 32) {
        float xv = xr[i];
        s1 += xv * ar[i];
        s2 += xv * br[i];
    }
    for (int off = 16; off > 0; off >>= 1) {
        s1 += __shfl_down(s1, off);
        s2 += __shfl_down(s2, off);
    }
    if (lane == 0) { asrc[pair] = s1; adst[pair] = s2; }
}

__global__ void fill_kernel(float* __restrict__ p, float v, int n)
{
    int i = blockIdx.x * blockDim.x + threadIdx.x;
    if (i < n) p[i] = v;
}

// float atomic max via signed/unsigned int ordering trick
__device__ inline void atomicMaxF(float* addr, float val)
{
    if (val >= 0.f) atomicMax((int*)addr, __float_as_int(val));
    else            atomicMin((unsigned int*)addr, (unsigned int)__float_as_int(val));
}

__global__ void edge_max_kernel(
    const float* __restrict__ asrc, const float* __restrict__ adst,
    const int* __restrict__ src, const int* __restrict__ dst,
    float* __restrict__ emax, int EH)
{
    int i = blockIdx.x * blockDim.x + threadIdx.x;
    if (i >= EH) return;
    int e = i >> 2, h = i & 3;
    float v = asrc[src[e] * 4 + h] + adst[dst[e] * 4 + h];
    v = v > 0.f ? v : 0.2f * v;                  // LeakyReLU(0.2)
    atomicMaxF(&emax[dst[e] * 4 + h], v);
}

__global__ void edge_exp_kernel(
    const float* __restrict__ asrc, const float* __restrict__ adst,
    const int* __restrict__ src, const int* __restrict__ dst,
    const float* __restrict__ emax, float* __restrict__ exb,
    float* __restrict__ denom, int EH)
{
    int i = blockIdx.x * blockDim.x + threadIdx.x;
    if (i >= EH) return;
    int e = i >> 2, h = i & 3;
    int d = dst[e];
    float v = asrc[src[e] * 4 + h] + adst[d * 4 + h];
    v = v > 0.f ? v : 0.2f * v;
    float ex = __expf(v - emax[d * 4 + h]);
    exb[i] = ex;
    atomicAdd(&denom[d * 4 + h], ex);
}

// One block per edge: scatter alpha * xl[src] into the accumulator (pre-loaded
// with the residual projection). concat=1: out[d,h*256+c]; concat=0: head-mean.
__global__ __launch_bounds__(256) void edge_aggregate_kernel(
    const float* __restrict__ xl, const float* __restrict__ exb,
    const float* __restrict__ denom, const int* __restrict__ src,
    const int* __restrict__ dst, float* __restrict__ out, int concat)
{
    int e = blockIdx.x;
    int c = threadIdx.x;
    int s = src[e], d = dst[e];
    const float* xr = xl + (size_t)s * 1024;
#pragma unroll
    for (int h = 0; h < 4; ++h) {
        float alpha = exb[e * 4 + h] / (denom[d * 4 + h] + 1e-16f);
        float m = alpha * xr[h * 256 + c];
        if (concat) atomicAdd(out + (size_t)d * 1024 + h * 256 + c, m);
        else        atomicAdd(out + (size_t)d * 256 + c, 0.25f * m);
    }
}

// Output heads: dep = h3@Wb+bb [N,2] ; sev = h3@Ws+bs [N,5]
__global__ void heads_kernel(
    const float* __restrict__ h3, const float* __restrict__ Wb,
    const float* __restrict__ bb, const float* __restrict__ Ws,
    const float* __restrict__ bs, float* __restrict__ out_dep,
    float* __restrict__ out_sev, int N)
{
    int idx = blockIdx.x * blockDim.x + threadIdx.x;
    if (idx >= N * 7) return;
    int n = idx / 7, j = idx % 7;
    const float* hr = h3 + (size_t)n * 256;
    float s = 0.f;
    if (j < 2) {
        for (int c = 0; c < 256; ++c) s += hr[c] * Wb[c * 2 + j];
        out_dep[n * 2 + j] = s + bb[j];
    } else {
        int k = j - 2;
        for (int c = 0; c < 256; ++c) s += hr[c] * Ws[c * 5 + k];
        out_sev[n * 5 + k] = s + bs[k];
    }
}

extern "C" void kernel_launch(void* const* d_in, const int* in_sizes, int n_in,
                              void* d_out, int out_size, void* d_ws, size_t ws_size,
                              hipStream_t stream)
{
    const float* x   = (const float*)d_in[0];
    const int*   ei  = (const int*)  d_in[1];
    const float* W1  = (const float*)d_in[2];
    const float* as1 = (const float*)d_in[3];
    const float* ad1 = (const float*)d_in[4];
    const float* Wr1 = (const float*)d_in[5];
    const float* b1  = (const float*)d_in[6];
    const float* W2  = (const float*)d_in[7];
    const float* as2 = (const float*)d_in[8];
    const float* ad2 = (const float*)d_in[9];
    const float* Wr2 = (const float*)d_in[10];
    const float* b2  = (const float*)d_in[11];
    const float* Wfc = (const float*)d_in[12];
    const float* bfc = (const float*)d_in[13];
    const float* Wb  = (const float*)d_in[14];
    const float* bb  = (const float*)d_in[15];
    const float* Ws  = (const float*)d_in[16];
    const float* bs  = (const float*)d_in[17];

    const int* src = ei;             // edge_index[0]
    const int* dst = ei + NEDGES;    // edge_index[1]

    // Workspace layout (fp32)
    float* ws   = (float*)d_ws;
    float* xl   = ws;                                 // N*1024 : xl1 -> xl2 -> (t,h3)
    float* h1   = xl  + (size_t)NNODES * 1024;        // N*1024 : res1 -> h1 accumulator
    float* h2   = h1  + (size_t)NNODES * 1024;        // N*256  : res2 -> h2 accumulator
    float* asrc = h2  + (size_t)NNODES * 256;         // N*4
    float* adst = asrc + NNODES * 4;                  // N*4
    float* emax = adst + NNODES * 4;                  // N*4
    float* den  = emax + NNODES * 4;                  // N*4
    float* exb  = den  + NNODES * 4;                  // E*4

    const int NH = NNODES * 4;
    const int EH = NEDGES * 4;
    dim3 blk(256);
    dim3 gN1024((NNODES + 127) / 128, 1024 / 64);
    dim3 gN256 ((NNODES + 127) / 128, 256 / 64);
    const float NEG_INF = -__builtin_huge_valf();

    // ---------------- Layer 1 (concat heads -> 1024) ----------------
    gemm_wmma_f32<<<gN1024, blk, 0, stream>>>(x, W1,  nullptr, xl, NNODES, 1024, 768, 0);
    gemm_wmma_f32<<<gN1024, blk, 0, stream>>>(x, Wr1, b1,      h1, NNODES, 1024, 768, 0);
    att_scores_kernel<<<(NH + 7) / 8, 256, 0, stream>>>(xl, as1, ad1, asrc, adst, NH);
    fill_kernel<<<(NH + 255) / 256, 256, 0, stream>>>(emax, NEG_INF, NH);
    fill_kernel<<<(NH + 255) / 256, 256, 0, stream>>>(den, 0.f, NH);
    edge_max_kernel<<<(EH + 255) / 256, 256, 0, stream>>>(asrc, adst, src, dst, emax, EH);
    edge_exp_kernel<<<(EH + 255) / 256, 256, 0, stream>>>(asrc, adst, src, dst, emax, exb, den, EH);
    edge_aggregate_kernel<<<NEDGES, 256, 0, stream>>>(xl, exb, den, src, dst, h1, 1);

    // ---------------- Layer 2 (mean heads -> 256) ----------------
    gemm_wmma_f32<<<gN1024, blk, 0, stream>>>(h1, W2,  nullptr, xl, NNODES, 1024, 1024, 0);
    gemm_wmma_f32<<<gN256,  blk, 0, stream>>>(h1, Wr2, b2,      h2, NNODES, 256, 1024, 0);
    att_scores_kernel<<<(NH + 7) / 8, 256, 0, stream>>>(xl, as2, ad2, asrc, adst, NH);
    fill_kernel<<<(NH + 255) / 256, 256, 0, stream>>>(emax, NEG_INF, NH);
    fill_kernel<<<(NH + 255) / 256, 256, 0, stream>>>(den, 0.f, NH);
    edge_max_kernel<<<(EH + 255) / 256, 256, 0, stream>>>(asrc, adst, src, dst, emax, EH);
    edge_exp_kernel<<<(EH + 255) / 256, 256, 0, stream>>>(asrc, adst, src, dst, emax, exb, den, EH);
    edge_aggregate_kernel<<<NEDGES, 256, 0, stream>>>(xl, exb, den, src, dst, h2, 0);

    // ---------------- FC chain + heads ----------------
    float* tbuf = xl;                               // xl2 no longer needed
    float* h3   = xl + (size_t)NNODES * 256;
    gemm_wmma_f32<<<gN256, blk, 0, stream>>>(h2,   Wfc, bfc, tbuf, NNODES, 256, 256, 1);
    gemm_wmma_f32<<<gN256, blk, 0, stream>>>(tbuf, Wfc, bfc, h3,   NNODES, 256, 256, 0);

    float* out_dep = (float*)d_out;
    float* out_sev = out_dep + (size_t)NNODES * 2;
    heads_kernel<<<(NNODES * 7 + 255) / 256, 256, 0, stream>>>(h3, Wb, bb, Ws, bs, out_dep, out_sev, NNODES);
}